// stgru_raw_84009560310374
// MI455X (gfx1250) — compile-verified
//
#include <hip/hip_runtime.h>

// Problem constants (from reference)
#define Bn 8
#define Hn 512
#define Wn 1024
#define HW (Hn*Wn)
#define TY 16
#define TX 64
#define S1 (TX+2)      // 66  : 1-halo stride
#define R1 (TY+2)      // 18
#define S2 (TX+4)      // 68  : 2-halo stride
#define R2 (TY+4)      // 20
#define NPIX1 (R1*S1)  // 1188
#define NPIX2 (R2*S2)  // 1360
#define HPR_OFF NPIX1
#define NT 256

#ifdef __HIP_DEVICE_COMPILE__
#if __has_builtin(__builtin_amdgcn_wmma_f32_16x16x4_f32)
#define USE_WMMA4 1
#endif
#endif
#ifndef USE_WMMA4
#define USE_WMMA4 0
#endif

typedef __attribute__((ext_vector_type(2))) float v2f;
typedef __attribute__((ext_vector_type(8))) float v8f;

// Fast reciprocal: v_rcp_f32 (~1 ulp), avoids the IEEE div expansion
__device__ __forceinline__ float frcp(float x) {
#ifdef __HIP_DEVICE_COMPILE__
  return __builtin_amdgcn_rcpf(x);
#else
  return 1.0f / x;
#endif
}

__device__ __forceinline__ float sigm(float x) { return frcp(1.0f + __expf(-x)); }

// Bilinear gather descriptor (zeros padding, align_corners=True semantics of ref)
struct G4 { int i0, i1, i2, i3; float w0, w1, w2, w3, sw; };

__device__ __forceinline__ G4 mkg(float py, float px) {
  G4 g;
  float y0 = floorf(py), x0 = floorf(px);
  float wy = py - y0, wx = px - x0;
  float y1 = y0 + 1.0f, x1 = x0 + 1.0f;
  bool vy0 = (y0 >= 0.f) && (y0 <= (float)(Hn - 1));
  bool vy1 = (y1 >= 0.f) && (y1 <= (float)(Hn - 1));
  bool vx0 = (x0 >= 0.f) && (x0 <= (float)(Wn - 1));
  bool vx1 = (x1 >= 0.f) && (x1 <= (float)(Wn - 1));
  int iy0 = min(max((int)y0, 0), Hn - 1), iy1 = min(max((int)y1, 0), Hn - 1);
  int ix0 = min(max((int)x0, 0), Wn - 1), ix1 = min(max((int)x1, 0), Wn - 1);
  g.w0 = (vy0 && vx0) ? (1.f - wy) * (1.f - wx) : 0.f;
  g.w1 = (vy0 && vx1) ? (1.f - wy) * wx : 0.f;
  g.w2 = (vy1 && vx0) ? wy * (1.f - wx) : 0.f;
  g.w3 = (vy1 && vx1) ? wy * wx : 0.f;
  g.sw = g.w0 + g.w1 + g.w2 + g.w3;
  g.i0 = iy0 * Wn + ix0; g.i1 = iy0 * Wn + ix1;
  g.i2 = iy1 * Wn + ix0; g.i3 = iy1 * Wn + ix1;
  return g;
}

// Combined weight matrix Wt(f,k): filters {0:xh,1:hh,2:xz,3:hz} x K=[0..8]=unary taps, [9..17]=hpr taps
__device__ __forceinline__ float wsel(int f, int k,
                                      const float* __restrict__ Wxh, const float* __restrict__ Whh,
                                      const float* __restrict__ Wxz, const float* __restrict__ Whz) {
  if (k < 9) {
    if (f == 0) return Wxh[k];
    if (f == 2) return Wxz[k];
    return 0.f;
  } else if (k < 18) {
    if (f == 1) return Whh[k - 9];
    if (f == 3) return Whz[k - 9];
    return 0.f;
  }
  return 0.f;
}

// LDS offset (in floats) for K-row k relative to pixel base in the unary||hpr union array.
// k>=18 -> safe dummy offset 0 (its weight row is all-zero so the product vanishes).
__device__ __forceinline__ int reloff(int k) {
  if (k >= 18) return 0;
  int base = 0, t = k;
  if (k >= 9) { base = HPR_OFF; t = k - 9; }
  int dy = t / 3, dx = t - 3 * dy;
  return base + dy * S1 + dx;
}

__global__ __launch_bounds__(NT) void stgru_fused_kernel(
    const float* __restrict__ img, const float* __restrict__ ref,
    const float* __restrict__ flow, const float* __restrict__ hprev,
    const float* __restrict__ unary,
    const float* __restrict__ Wir, const float* __restrict__ Wxh,
    const float* __restrict__ Whh, const float* __restrict__ Wxz,
    const float* __restrict__ Whz, const float* __restrict__ lambp,
    const float* __restrict__ brp, const float* __restrict__ bzp,
    float* __restrict__ out) {
  // LDS: I_diff (3ch, 2-halo) + union [unary(1-halo) | h_prev_reset(1-halo)]
  __shared__ float sId[3 * NPIX2];
  __shared__ float sU[2 * NPIX1];

  const int tid = threadIdx.x;
  const int bx = blockIdx.x, by = blockIdx.y, b = blockIdx.z;
  const int y0t = by * TY, x0t = bx * TX;

  // ImageNet normalization: w = 1/std, b = -mean/std
  const float nw0 = 1.f / 0.229f, nw1 = 1.f / 0.224f, nw2 = 1.f / 0.225f;
  const float nb0 = -0.485f / 0.229f, nb1 = -0.456f / 0.224f, nb2 = -0.406f / 0.225f;

  const float* imgb = img + (size_t)b * 3 * HW;
  const float* refb = ref + (size_t)b * 3 * HW;
  const float* fl0 = flow + (size_t)b * 2 * HW;
  const float* fl1 = fl0 + HW;
  const float* hpb = hprev + (size_t)b * HW;
  const float* unb = unary + (size_t)b * HW;

  // ---- Stage 1a: I_diff = norm(image) - warp(norm(image_ref)) on 2-halo region
  for (int p = tid; p < NPIX2; p += NT) {
    int ly = p / S2, lx = p - ly * S2;
    int gy = y0t + ly - 2, gx = x0t + lx - 2;
    float d0 = 0.f, d1 = 0.f, d2 = 0.f;
    if ((unsigned)gy < (unsigned)Hn && (unsigned)gx < (unsigned)Wn) {
      int gi = gy * Wn + gx;
      G4 g = mkg((float)gy + fl0[gi], (float)gx + fl1[gi]);
      const float* r0 = refb;
      const float* r1 = refb + HW;
      const float* r2 = refb + 2 * HW;
      float s0 = g.w0 * r0[g.i0] + g.w1 * r0[g.i1] + g.w2 * r0[g.i2] + g.w3 * r0[g.i3];
      float s1 = g.w0 * r1[g.i0] + g.w1 * r1[g.i1] + g.w2 * r1[g.i2] + g.w3 * r1[g.i3];
      float s2 = g.w0 * r2[g.i0] + g.w1 * r2[g.i1] + g.w2 * r2[g.i2] + g.w3 * r2[g.i3];
      float oms = 1.f - g.sw;  // affine bias only where warp has support
      d0 = nw0 * (imgb[gi] - s0) + nb0 * oms;
      d1 = nw1 * (imgb[HW + gi] - s1) + nb1 * oms;
      d2 = nw2 * (imgb[2 * HW + gi] - s2) + nb2 * oms;
    }
    sId[p] = d0; sId[NPIX2 + p] = d1; sId[2 * NPIX2 + p] = d2;
  }

  // ---- Stage 1b: sigmoid(unary)-0.5 and warp(sigmoid(h_prev)-0.5) on 1-halo region
  for (int p = tid; p < NPIX1; p += NT) {
    int ly = p / S1, lx = p - ly * S1;
    int gy = y0t + ly - 1, gx = x0t + lx - 1;
    float uv = 0.f, hw = 0.f;
    if ((unsigned)gy < (unsigned)Hn && (unsigned)gx < (unsigned)Wn) {
      int gi = gy * Wn + gx;
      uv = sigm(unb[gi]) - 0.5f;
      G4 g = mkg((float)gy + fl0[gi], (float)gx + fl1[gi]);
      hw = g.w0 * sigm(hpb[g.i0]) + g.w1 * sigm(hpb[g.i1]) +
           g.w2 * sigm(hpb[g.i2]) + g.w3 * sigm(hpb[g.i3]) - 0.5f * g.sw;
    }
    sU[p] = uv;
    sU[HPR_OFF + p] = hw;
  }
  __syncthreads();

  // ---- Stage 2: r = 1 - tanh(|conv(I_diff,W_ir)+bias_r|); h_prev_reset = h_prev_warped * r
  float wir[27];
#pragma unroll
  for (int i = 0; i < 27; ++i) wir[i] = Wir[i];
  const float biasr = brp[0];
  for (int p = tid; p < NPIX1; p += NT) {
    int iy = p / S1, ix = p - iy * S1;
    float acc = biasr;
#pragma unroll
    for (int ch = 0; ch < 3; ++ch) {
      const float* sc = &sId[ch * NPIX2 + iy * S2 + ix];
#pragma unroll
      for (int dy = 0; dy < 3; ++dy)
#pragma unroll
        for (int dx = 0; dx < 3; ++dx)
          acc += sc[dy * S2 + dx] * wir[ch * 9 + dy * 3 + dx];
    }
    float t = __expf(-2.f * fabsf(acc));
    float r = 2.f * t * frcp(1.f + t);  // == 1 - tanh(|acc|)
    sU[HPR_OFF + p] *= r;
  }
  __syncthreads();

  // ---- Stage 3: four 9-tap convs as D(4x16) = Wt(4x18) x taps(18x16) via V_WMMA_F32_16X16X4_F32
  const float lamb = lambp[0];
  const float biasz = bzp[0];
  const int lane = tid & 31;
  const int wv = tid >> 5;     // 8 wave32s
  const int f = lane & 15;     // filter row (A) / pixel column (B)
  const int g = lane >> 4;     // K-half within chunk

  v2f aw[5];
  int ro0[5], ro1[5];
#pragma unroll
  for (int c = 0; c < 5; ++c) {
    int k = 4 * c + 2 * g;  // A & B: VGPR j of lane-group g holds K = k0 + 2g + j
    aw[c].x = wsel(f, k, Wxh, Whh, Wxz, Whz);
    aw[c].y = wsel(f, k + 1, Wxh, Whh, Wxz, Whz);
    ro0[c] = reloff(k);
    ro1[c] = reloff(k + 1);
  }

#pragma unroll
  for (int i = 0; i < 8; ++i) {
    int ch = wv * 8 + i;           // 64 chunks: 16 rows x 4 x-groups of 16 pixels
    int yl = ch >> 2;
    int xl0 = (ch & 3) * 16;
    int pixBase = yl * S1 + xl0 + f;  // 1-halo tap base for pixel column f
    float yxh, yhh, yxz, yhz;
#if USE_WMMA4
    v8f acc = {0.f, 0.f, 0.f, 0.f, 0.f, 0.f, 0.f, 0.f};
#pragma unroll
    for (int c = 0; c < 5; ++c) {
      v2f bm;
      bm.x = sU[pixBase + ro0[c]];
      bm.y = sU[pixBase + ro1[c]];
      // (neg_a, A, neg_b, B, c_mod, C, reuse_a, reuse_b)
      acc = __builtin_amdgcn_wmma_f32_16x16x4_f32(false, aw[c], false, bm,
                                                  (short)0, acc, false, false);
    }
    // lanes 0-15: D columns N=lane (pixel), rows M=0..3 (filters) in acc.s0..s3
    yxh = acc.s0; yhh = acc.s1; yxz = acc.s2; yhz = acc.s3;
#else
    yxh = 0.f; yhh = 0.f; yxz = 0.f; yhz = 0.f;
    if (lane < 16) {
      int pb = yl * S1 + xl0 + lane;
#pragma unroll
      for (int t = 0; t < 9; ++t) {
        int dy = t / 3, dx = t - 3 * dy;
        float u = sU[pb + dy * S1 + dx];
        float hp = sU[HPR_OFF + pb + dy * S1 + dx];
        yxh += u * Wxh[t]; yxz += u * Wxz[t];
        yhh += hp * Whh[t]; yhz += hp * Whz[t];
      }
    }
#endif
    if (lane < 16) {
      float htl = yxh + yhh;
      float z = sigm(yxz + yhz + biasz);
      float hc = sU[HPR_OFF + (yl + 1) * S1 + (xl0 + lane + 1)];
      float h = lamb * (1.f - z) * hc + z * htl;
      int gy = y0t + yl, gx = x0t + xl0 + lane;
      out[(size_t)b * HW + (size_t)gy * Wn + gx] = h;
    }
  }
}

extern "C" void kernel_launch(void* const* d_in, const int* in_sizes, int n_in,
                              void* d_out, int out_size, void* d_ws, size_t ws_size,
                              hipStream_t stream) {
  (void)in_sizes; (void)n_in; (void)d_ws; (void)ws_size; (void)out_size;
  dim3 grid(Wn / TX, Hn / TY, Bn);  // 16 x 32 x 8 = 4096 workgroups
  dim3 block(NT);
  stgru_fused_kernel<<<grid, block, 0, stream>>>(
      (const float*)d_in[0],  // image
      (const float*)d_in[1],  // image_ref
      (const float*)d_in[2],  // flow
      (const float*)d_in[3],  // h_prev
      (const float*)d_in[4],  // unary_input
      (const float*)d_in[5],  // W_ir
      (const float*)d_in[6],  // W_hh  -> actually W_xh (see order below)
      (const float*)d_in[7],  // W_hh
      (const float*)d_in[8],  // W_xz
      (const float*)d_in[9],  // W_hz
      (const float*)d_in[10], // lamb
      (const float*)d_in[11], // bias_r
      (const float*)d_in[12], // bias_z
      (float*)d_out);
}